// GetDensity_77962246357628
// MI455X (gfx1250) — compile-verified
//
#include <hip/hip_runtime.h>
#include <math.h>

typedef float v2f __attribute__((ext_vector_type(2)));
typedef float v8f __attribute__((ext_vector_type(8)));

// 4-way register select: s in [0,3] -> a,b,c,d (3 cndmasks)
__device__ __forceinline__ float sel4(int s, float a, float b, float c, float d) {
  float lo = (s & 1) ? b : a;
  float hi = (s & 1) ? d : c;
  return (s & 2) ? hi : lo;
}

// ---------------- bucket construction (counting sort by center atom) ----------------

__global__ void count_kernel(const int* __restrict__ i0, unsigned* __restrict__ counts, int nneigh) {
  int n = blockIdx.x * blockDim.x + threadIdx.x;
  if (n < nneigh) atomicAdd(&counts[i0[n]], 1u);
}

// single-block Hillis-Steele scan with running carry: offsets[0..n] = exclusive scan of counts
__global__ void scan_kernel(const unsigned* __restrict__ counts, unsigned* __restrict__ offsets, int n) {
  __shared__ unsigned sm[1024];
  const int t = threadIdx.x;
  unsigned carry = 0;
  if (t == 0) offsets[0] = 0;
  for (int base = 0; base < n; base += 1024) {
    const int i = base + t;
    sm[t] = (i < n) ? counts[i] : 0u;
    __syncthreads();
    for (int d = 1; d < 1024; d <<= 1) {
      unsigned add = (t >= d) ? sm[t - d] : 0u;
      __syncthreads();
      sm[t] += add;
      __syncthreads();
    }
    if (i < n) offsets[i + 1] = sm[t] + carry;
    unsigned tot = sm[1023];
    __syncthreads();
    carry += tot;
  }
}

// store packed record (species<<20 | i1) so the hot loop has a 2-level (not 3-level) chase
__global__ void scatter_kernel(const int* __restrict__ i0, const int* __restrict__ i1,
                               const int* __restrict__ species,
                               const unsigned* __restrict__ offsets,
                               unsigned* __restrict__ cursor, unsigned* __restrict__ packed,
                               int nneigh) {
  int n = blockIdx.x * blockDim.x + threadIdx.x;
  if (n < nneigh) {
    int c = i0[n];
    unsigned pos = atomicAdd(&cursor[c], 1u);
    packed[offsets[c] + pos] = (unsigned)i1[n] | ((unsigned)species[n] << 20);
  }
}

// ---------------- main density kernel: one wave32 per center atom ----------------
// sumw_tile(16x16) = Angular(13xK) x Radial(Kx16), 4 neighbors per
// V_WMMA_F32_16X16X4_F32, then per-row-group sum of squares -> out[i, 3, 16].

__global__ void __launch_bounds__(256)
density_kernel(const float* __restrict__ cart, const float* __restrict__ rs,
               const float* __restrict__ inta, const float* __restrict__ params,
               const unsigned* __restrict__ counts, const unsigned* __restrict__ offsets,
               const unsigned* __restrict__ packed, float* __restrict__ out, int nlocal) {
  const int wave = blockIdx.x * (blockDim.x >> 5) + (threadIdx.x >> 5);
  const int lane = threadIdx.x & 31;
  if (wave >= nlocal) return;                 // whole-wave uniform; EXEC full below

  const int t    = lane & 15;                 // angular row j == radial channel k
  const int half = lane >> 4;

  const int i = wave;
  const unsigned ni   = counts[i];
  const unsigned base = offsets[i];
  const float cx = cart[3 * i + 0];
  const float cy = cart[3 * i + 1];
  const float cz = cart[3 * i + 2];

  // species tables resident in registers (channel t): rs/inta/params for s=0..3
  const float r0 = rs[t],     r1 = rs[16 + t],     r2 = rs[32 + t],     r3 = rs[48 + t];
  const float q0 = inta[t],   q1 = inta[16 + t],   q2 = inta[32 + t],   q3 = inta[48 + t];
  const float p0 = params[t], p1 = params[16 + t], p2 = params[32 + t], p3 = params[48 + t];

  // branchless angular row t: fc * dsel[ia2] * dsel[ib2] * zmask  (slot 3 == 1.0)
  const int   ia2   = (t == 0) ? 3 : ((t <= 3) ? (t - 1) : (t - 4) / 3);
  const int   ib2   = (t <= 3) ? 3 : (t - 4) % 3;
  const float zmask = (t <= 12) ? 1.0f : 0.0f;

  auto eval = [&](unsigned pk, float& ang, float& rad) {
    const int i1 = (int)(pk & 0xFFFFFu);
    const int s  = (int)(pk >> 20);
    const float dx = cx - cart[3 * i1 + 0];
    const float dy = cy - cart[3 * i1 + 1];
    const float dz = cz - cart[3 * i1 + 2];
    const float rsq  = dx * dx + dy * dy + dz * dz;
    const float inv  = __builtin_amdgcn_rsqf(rsq);      // v_rsq_f32
    const float dist = rsq * inv;                       // sqrt(rsq)
    float fc = 0.5f * __cosf(dist * 0.62831853071795864769f) + 0.5f;  // pi/CUTOFF
    fc *= fc;
    const float d0 = dx * inv, d1 = dy * inv, d2 = dz * inv;
    ang = fc * zmask * sel4(ia2, d0, d1, d2, 1.0f) * sel4(ib2, d0, d1, d2, 1.0f);
    const float rr = sel4(s, r0, r1, r2, r3);
    const float qq = sel4(s, q0, q1, q2, q3);
    const float pp = sel4(s, p0, p1, p2, p3);
    const float dd = dist - rr;
    rad = pp * __expf(-qq * dd * dd);                   // v_exp_f32; params folded in
  };

  v8f acc = {0.f, 0.f, 0.f, 0.f, 0.f, 0.f, 0.f, 0.f};

  // branch-free steady state: 4 neighbors per WMMA (K-slots), 2 per lane
  const unsigned nfull = ni & ~3u;
  unsigned cb = 0;
  for (; cb < nfull; cb += 4) {
    const unsigned pkX = packed[base + cb + 2u * (unsigned)half];
    const unsigned pkY = packed[base + cb + 2u * (unsigned)half + 1u];
    float aX, rX, aY, rY;
    eval(pkX, aX, rX);
    eval(pkY, aY, rY);
    v2f A = {aX, aY};
    v2f B = {rX, rY};
    acc = __builtin_amdgcn_wmma_f32_16x16x4_f32(false, A, false, B, (short)0, acc, false, false);
  }

  if (cb < ni) {  // wave-uniform tail chunk; zero-padded slots contribute nothing
    const unsigned ix = cb + 2u * (unsigned)half;
    const unsigned iy = ix + 1u;
    float aX = 0.f, rX = 0.f, aY = 0.f, rY = 0.f;
    if (ix < ni) eval(packed[base + ix], aX, rX);
    if (iy < ni) eval(packed[base + iy], aY, rY);
    v2f A = {aX, aY};
    v2f B = {rX, rY};
    acc = __builtin_amdgcn_wmma_f32_16x16x4_f32(false, A, false, B, (short)0, acc, false, false);
  }

  // per-lane: column n = t; half 0 holds rows 0..7, half 1 holds rows 8..15 (13..15 exact zeros)
  float s0 = 0.f, s1 = 0.f, s2p = 0.f;
  if (half == 0) {
    s0  = acc[0] * acc[0];
    s1  = acc[1] * acc[1] + acc[2] * acc[2] + acc[3] * acc[3];
    s2p = acc[4] * acc[4] + acc[5] * acc[5] + acc[6] * acc[6] + acc[7] * acc[7];
  } else {
    s2p = acc[0] * acc[0] + acc[1] * acc[1] + acc[2] * acc[2] + acc[3] * acc[3] + acc[4] * acc[4];
  }
  const float s2 = s2p + __shfl_xor(s2p, 16, 32);  // rows 4..7 + rows 8..12

  if (half == 0) {
    float* o = out + (size_t)i * 48;
    o[t]      = s0;   // order 0
    o[16 + t] = s1;   // order 1
    o[32 + t] = s2;   // order 2
  }
}

// ---------------- launch ----------------

extern "C" void kernel_launch(void* const* d_in, const int* in_sizes, int n_in,
                              void* d_out, int out_size, void* d_ws, size_t ws_size,
                              hipStream_t stream) {
  const float* cart          = (const float*)d_in[0];
  const float* rs            = (const float*)d_in[1];
  const float* inta          = (const float*)d_in[2];
  const float* params        = (const float*)d_in[3];
  const int*   atom_index    = (const int*)d_in[4];
  const int*   neigh_species = (const int*)d_in[6];
  const int nlocal = in_sizes[5];
  const int nneigh = in_sizes[6];
  const int* i0 = atom_index;
  const int* i1 = atom_index + nneigh;

  char* ws = (char*)d_ws;
  unsigned* counts  = (unsigned*)(ws);                        // nlocal u32
  unsigned* cursor  = (unsigned*)(ws + (size_t)nlocal * 4);   // nlocal u32
  unsigned* offsets = (unsigned*)(ws + (size_t)nlocal * 8);   // nlocal+1 u32
  unsigned* packed  = (unsigned*)(ws + (size_t)nlocal * 12 + 256);  // nneigh u32

  hipMemsetAsync(counts, 0, (size_t)nlocal * 8, stream);      // counts + cursor

  const int tb = 256;
  count_kernel  <<<(nneigh + tb - 1) / tb, tb, 0, stream>>>(i0, counts, nneigh);
  scan_kernel   <<<1, 1024, 0, stream>>>(counts, offsets, nlocal);
  scatter_kernel<<<(nneigh + tb - 1) / tb, tb, 0, stream>>>(i0, i1, neigh_species,
                                                            offsets, cursor, packed, nneigh);

  const int waves_per_block = tb / 32;                                  // 8
  const int blocks = (nlocal + waves_per_block - 1) / waves_per_block;  // 5000
  density_kernel<<<blocks, tb, 0, stream>>>(cart, rs, inta, params,
                                            counts, offsets, packed, (float*)d_out, nlocal);
}